// CameraMemory_42640435314783
// MI455X (gfx1250) — compile-verified
//
#include <hip/hip_runtime.h>
#include <math.h>

#define DDIM 256
#define INV_TEMP 20.0f   // 1 / 0.05
#define BM 64
#define BN 128
#define BK 32
#define PADW 36          // LDS row stride in floats: 144B (16B aligned), conflict-free

typedef float v2f  __attribute__((ext_vector_type(2)));
typedef float v4fx __attribute__((ext_vector_type(4)));
typedef float v8f  __attribute__((ext_vector_type(8)));

// Types matching the async-LDS builtin prototype:
//   void __builtin_amdgcn_global_load_async_to_lds_b128(v4i AS1*, v4i AS3*, imm, imm)
typedef int v4i __attribute__((vector_size(4 * sizeof(int))));
typedef __attribute__((address_space(1))) v4i* gbl_v4i_p;
typedef __attribute__((address_space(3))) v4i* lds_v4i_p;

// ---- gfx1250 async global->LDS path (guarded; falls back to sync copy) ----
#if defined(__AMDGCN__) && __has_builtin(__builtin_amdgcn_global_load_async_to_lds_b128)
#define USE_ASYNC_LDS 1
#else
#define USE_ASYNC_LDS 0
#endif

__device__ __forceinline__ void copy16_g2l(const float* __restrict__ g, float* l) {
#if USE_ASYNC_LDS
  // Direct global -> LDS DMA, tracked by ASYNCcnt; no VGPR staging.
  __builtin_amdgcn_global_load_async_to_lds_b128((gbl_v4i_p)g, (lds_v4i_p)l, 0, 0);
#else
  *(v4fx*)l = *(const v4fx*)g;
#endif
}

__device__ __forceinline__ void async_wait_all() {
#if USE_ASYNC_LDS
#if __has_builtin(__builtin_amdgcn_s_wait_asynccnt)
  __builtin_amdgcn_s_wait_asynccnt(0);
#else
  asm volatile("s_wait_asynccnt 0" ::: "memory");
#endif
#endif
}

// ---------------------------------------------------------------------------
// Kernel 1: row-wise L2 normalize of inputs [B, 256] -> xn (workspace)
// ---------------------------------------------------------------------------
__global__ __launch_bounds__(256) void normalize_rows(const float* __restrict__ x,
                                                      float* __restrict__ xn) {
  __shared__ float red[256];
  const int row = blockIdx.x;
  const int t = threadIdx.x;
  float v = x[row * DDIM + t];
  red[t] = v * v;
  __syncthreads();
  for (int s = 128; s > 0; s >>= 1) {
    if (t < s) red[t] += red[t + s];
    __syncthreads();
  }
  const float scale = 1.0f / fmaxf(sqrtf(red[0]), 1e-12f);
  xn[row * DDIM + t] = v * scale;
}

// ---------------------------------------------------------------------------
// Kernel 2: fused  (xn @ feat^T -> s -> circle-loss terms -> block partials)
// Block: 256 threads = 8 waves (2x4).  Block tile: 64(M) x 128(N), K-chunk 32.
// Each wave: 32x32 output = four 16x16 f32 WMMA accumulators.
// ---------------------------------------------------------------------------
__global__ __launch_bounds__(256) void gemm_circle_loss(
    const float* __restrict__ xn, const float* __restrict__ feat,
    const int* __restrict__ targets, const int* __restrict__ cams,
    const int* __restrict__ pids, const int* __restrict__ camids,
    float* __restrict__ partials, int Ntot) {
  __shared__ __align__(16) float As[2][BM][PADW];
  __shared__ __align__(16) float Bs[2][BN][PADW];
  __shared__ float redp[256];
  __shared__ float redn[256];

  const int tid = threadIdx.x;
  const int lane = tid & 31;
  const int w = tid >> 5;          // wave id 0..7
  const int wm = w >> 2;           // 0..1  (M wave coord)
  const int wn = w & 3;            // 0..3  (N wave coord)
  const int laneHalf = lane >> 4;  // 0/1
  const int l16 = lane & 15;

  const int mBase = blockIdx.y * BM;
  const int nBase = blockIdx.x * BN;

  v8f acc[2][2];
#pragma unroll
  for (int tm = 0; tm < 2; ++tm)
#pragma unroll
    for (int tn = 0; tn < 2; ++tn) acc[tm][tn] = v8f{};

  // ---- tile loaders (global -> LDS, async on gfx1250) ----
  auto loadA = [&](int buf, int kBase) {
#pragma unroll
    for (int j = 0; j < 2; ++j) {        // 64x32 floats = 512 float4, 2/thread
      const int i = tid + j * 256;
      const int r = i >> 3;
      const int c = (i & 7) << 2;
      copy16_g2l(xn + (mBase + r) * DDIM + kBase + c, &As[buf][r][c]);
    }
  };
  auto loadB = [&](int buf, int kBase) {
#pragma unroll
    for (int j = 0; j < 4; ++j) {        // 128x32 floats = 1024 float4, 4/thread
      const int i = tid + j * 256;
      const int r = i >> 3;
      const int c = (i & 7) << 2;
      const int n = nBase + r;
      if (n < Ntot) {
        const float* gp = feat + (long)n * DDIM + kBase + c;
        copy16_g2l(gp, &Bs[buf][r][c]);
        if (kBase + BK < DDIM) __builtin_prefetch(gp + BK, 0, 3);  // global_prefetch_b8
      } else {
        *(v4fx*)&Bs[buf][r][c] = v4fx{};   // zero-fill OOB rows (epilogue masks them too)
      }
    }
  };

  // ---- per-chunk compute: 8 K-steps of 4, 4 WMMAs per step per wave ----
  auto compute = [&](int buf) {
#pragma unroll
    for (int kk = 0; kk < BK; kk += 4) {
      // fp32 16x4 A-fragment layout: per lane {K=kb, K=kb+1}, kb = kk + 2*laneHalf
      const int kb = kk + (laneHalf << 1);
      v2f af[2], bf[2];
#pragma unroll
      for (int tm = 0; tm < 2; ++tm) {
        const int r = wm * 32 + tm * 16 + l16;
        af[tm].x = As[buf][r][kb];
        af[tm].y = As[buf][r][kb + 1];
      }
#pragma unroll
      for (int tn = 0; tn < 2; ++tn) {
        const int c = wn * 32 + tn * 16 + l16;
        bf[tn].x = Bs[buf][c][kb];   // B[k][n] = features[n][k]
        bf[tn].y = Bs[buf][c][kb + 1];
      }
#pragma unroll
      for (int tm = 0; tm < 2; ++tm)
#pragma unroll
        for (int tn = 0; tn < 2; ++tn)
          acc[tm][tn] = __builtin_amdgcn_wmma_f32_16x16x4_f32(
              false, af[tm], false, bf[tn], (short)0, acc[tm][tn], false, false);
    }
  };

  // ---- double-buffered main loop over K ----
  int buf = 0;
  loadA(0, 0);
  loadB(0, 0);
  async_wait_all();
  __syncthreads();
  for (int kc = 0; kc < DDIM / BK; ++kc) {
    if (kc + 1 < DDIM / BK) {
      loadA(buf ^ 1, (kc + 1) * BK);   // async transfers overlap the WMMAs below
      loadB(buf ^ 1, (kc + 1) * BK);
    }
    compute(buf);
    async_wait_all();                  // next tile landed in LDS
    __syncthreads();                   // ...and is visible to all waves
    buf ^= 1;
  }

  // ---- fused circle-loss epilogue ----
  float lp = 0.0f, ln = 0.0f;
#pragma unroll
  for (int tm = 0; tm < 2; ++tm) {
#pragma unroll
    for (int tn = 0; tn < 2; ++tn) {
      const int n = nBase + wn * 32 + tn * 16 + l16;
      if (n < Ntot) {
        const int pid = pids[n];
        const int cid = camids[n];
#pragma unroll
        for (int vv = 0; vv < 8; ++vv) {
          // C/D layout: row = vv + 8*laneHalf within 16x16 tile, col = l16
          const int b = mBase + wm * 32 + tm * 16 + vv + laneHalf * 8;
          const float s = (acc[tm][tn][vv] + 1.0f) * 0.5f;
          const int tgt = targets[b];
          const int cam = cams[b];
          const bool pos = (pid == tgt);
          const bool neg = (!pos) && (cid == cam);
          const float ap = fmaxf(1.0f - s, 0.0f);
          const float an = fmaxf(s, 0.0f);
          if (pos) lp += expf(ap * (1.0f - s) * INV_TEMP);  // == exp(-ap*(s-1)/T)
          if (neg) ln += expf(an * s * INV_TEMP);
        }
      }
    }
  }

  // ---- block reduction, write deterministic per-block partials ----
  redp[tid] = lp;
  redn[tid] = ln;
  __syncthreads();
  for (int s = 128; s > 0; s >>= 1) {
    if (tid < s) {
      redp[tid] += redp[tid + s];
      redn[tid] += redn[tid + s];
    }
    __syncthreads();
  }
  if (tid == 0) {
    const int bi = blockIdx.y * gridDim.x + blockIdx.x;
    partials[2 * bi + 0] = redp[0];
    partials[2 * bi + 1] = redn[0];
  }
}

// ---------------------------------------------------------------------------
// Kernel 3: deterministic final reduction + log1p(lp * ln)
// ---------------------------------------------------------------------------
__global__ __launch_bounds__(256) void finalize_loss(const float* __restrict__ partials,
                                                     int nParts, float* __restrict__ out) {
  __shared__ float rp[256];
  __shared__ float rn[256];
  const int t = threadIdx.x;
  float sp = 0.0f, sn = 0.0f;
  for (int i = t; i < nParts; i += 256) {
    sp += partials[2 * i + 0];
    sn += partials[2 * i + 1];
  }
  rp[t] = sp;
  rn[t] = sn;
  __syncthreads();
  for (int s = 128; s > 0; s >>= 1) {
    if (t < s) {
      rp[t] += rp[t + s];
      rn[t] += rn[t + s];
    }
    __syncthreads();
  }
  if (t == 0) out[0] = log1pf(rp[0] * rn[0]);
}

// ---------------------------------------------------------------------------
extern "C" void kernel_launch(void* const* d_in, const int* in_sizes, int n_in,
                              void* d_out, int out_size, void* d_ws, size_t ws_size,
                              hipStream_t stream) {
  const float* inputs  = (const float*)d_in[0];
  const float* feats   = (const float*)d_in[1];
  const int*   targets = (const int*)d_in[2];
  const int*   cams    = (const int*)d_in[3];
  const int*   pids    = (const int*)d_in[4];
  const int*   camids  = (const int*)d_in[5];

  const int Brows = in_sizes[0] / DDIM;   // 256
  const int Ntot  = in_sizes[1] / DDIM;   // 100000

  float* xn       = (float*)d_ws;
  float* partials = (float*)((char*)d_ws + (size_t)Brows * DDIM * sizeof(float));

  normalize_rows<<<Brows, 256, 0, stream>>>(inputs, xn);

  dim3 grid((Ntot + BN - 1) / BN, Brows / BM);
  gemm_circle_loss<<<grid, 256, 0, stream>>>(xn, feats, targets, cams, pids, camids,
                                             partials, Ntot);

  const int nParts = grid.x * grid.y;
  finalize_loss<<<1, 256, 0, stream>>>(partials, nParts, (float*)d_out);
}